// BasePhotometric_2353642078741
// MI455X (gfx1250) — compile-verified
//
#include <hip/hip_runtime.h>
#include <hip/hip_bf16.h>

typedef __attribute__((ext_vector_type(2))) float v2f;
typedef __attribute__((ext_vector_type(8))) float v8f;

#define BB    8
#define CCH   3
#define HH    512
#define WW    1024
#define NBF   2
#define NPART 2048
#define PI_F  3.14159265358979323846f

// ---------------- zero partials ----------------
__global__ void zero_kernel(float* __restrict__ p) {
    p[blockIdx.x * blockDim.x + threadIdx.x] = 0.0f;
}

// ---------------- axis-angle -> R,t (16 of them) ----------------
__global__ void rot_kernel(const float* __restrict__ pose, float* __restrict__ rt) {
    int i = threadIdx.x;                      // 0..15 = (b*NBF + n)
    if (i >= BB * NBF) return;
    const float* p = pose + i * 6;
    float tx = p[0], ty = p[1], tz = p[2];
    float rx = p[3], ry = p[4], rz = p[5];
    float th  = sqrtf(rx * rx + ry * ry + rz * rz);
    float inv = 1.0f / fmaxf(th, 1e-8f);
    float kx = rx * inv, ky = ry * inv, kz = rz * inv;
    float s = sinf(th), c = cosf(th), mc = 1.0f - c;
    float* o = rt + i * 12;
    o[0] = 1.0f + mc * (-(ky * ky + kz * kz));
    o[1] = -s * kz + mc * kx * ky;
    o[2] =  s * ky + mc * kx * kz;
    o[3] =  s * kz + mc * kx * ky;
    o[4] = 1.0f + mc * (-(kx * kx + kz * kz));
    o[5] = -s * kx + mc * ky * kz;
    o[6] = -s * ky + mc * kx * kz;
    o[7] =  s * kx + mc * ky * kz;
    o[8] = 1.0f + mc * (-(kx * kx + ky * ky));
    o[9] = tx; o[10] = ty; o[11] = tz;
}

// ---------------- area downsample (direct from full res) ----------------
template <int S>
__device__ __forceinline__ float area_mean(const float* __restrict__ chan_full, int y, int x) {
    const float* src = chan_full + (size_t)(y * S) * WW + (size_t)(x * S);
    float acc = 0.0f;
#pragma unroll
    for (int dy = 0; dy < S; ++dy)
#pragma unroll
        for (int dx = 0; dx < S; ++dx)
            acc += src[(size_t)dy * WW + dx];
    return acc * (1.0f / (S * S));
}

template <int S>
__global__ void down_kernel(const float* __restrict__ in, float* __restrict__ out,
                            int h, int w) {
    int idx = blockIdx.x * blockDim.x + threadIdx.x;
    int total = BB * CCH * h * w;
    if (idx >= total) return;
    int x  = idx % w;
    int t1 = idx / w;
    int y  = t1 % h;
    int bc = t1 / h;
    out[idx] = area_mean<S>(in + (size_t)bc * HH * WW, y, x);
}

// ---------------- main photometric kernel per level ----------------
template <int LVL, bool PYR>
__global__ void photo_kernel(const float* __restrict__ ref,
                             const float* __restrict__ depth,
                             const float* __restrict__ mask,
                             const float* __restrict__ tgt0_full,
                             const float* __restrict__ tgt1_full,
                             const float* __restrict__ tgt0_lvl,
                             const float* __restrict__ tgt1_lvl,
                             const float* __restrict__ rt,
                             float* __restrict__ partials) {
    constexpr int S = 1 << LVL;
    constexpr int h = HH >> LVL;
    constexpr int w = WW >> LVL;

    int idx = blockIdx.x * 256 + threadIdx.x;   // B*h*w is a multiple of 256
    int x  = idx % w;
    int t1 = idx / w;
    int y  = t1 % h;
    int b  = t1 / h;

    // equirect direction for (y, x)
    float lon = ((x + 0.5f) / (float)w * 2.0f - 1.0f) * PI_F;
    float lat = -((y + 0.5f) / (float)h * 2.0f - 1.0f) * (0.5f * PI_F);
    float cl  = cosf(lat);
    float dxv = cl * sinf(lon);
    float dyv = sinf(lat);
    float dzv = cl * cosf(lon);

    float d = depth[((size_t)b * h + y) * w + x];
    float Xp = d * dxv, Yp = d * dyv, Zp = d * dzv;

    // downsampled ref pixel (on the fly; cheap, sequential reads)
    float rr[3];
#pragma unroll
    for (int c = 0; c < 3; ++c) {
        const float* chan = ref + (size_t)(b * 3 + c) * HH * WW;
        if (LVL == 0) rr[c] = chan[(size_t)y * WW + x];
        else          rr[c] = area_mean<S>(chan, y, x);
    }

    const float scale = 1.0f / ((float)BB * 3.0f * (float)h * (float)w);
    float lsum = 0.0f;

#pragma unroll
    for (int j = 0; j < NBF; ++j) {
        const float* Rt = rt + (size_t)(b * NBF + j) * 12;
        float px = Rt[0] * Xp + Rt[1] * Yp + Rt[2] * Zp + Rt[9];
        float py = Rt[3] * Xp + Rt[4] * Yp + Rt[5] * Zp + Rt[10];
        float pz = Rt[6] * Xp + Rt[7] * Yp + Rt[8] * Zp + Rt[11];
        float nrm = sqrtf(px * px + py * py + pz * pz);
        float lo  = atan2f(px, pz);
        float la  = asinf(fminf(fmaxf(py / fmaxf(nrm, 1e-8f), -1.0f), 1.0f));
        float gx  = lo / PI_F;
        float gy  = la / (0.5f * PI_F);

        float fx  = (gx + 1.0f) * 0.5f * (float)(w - 1);
        float fy  = (gy + 1.0f) * 0.5f * (float)(h - 1);
        float x0f = floorf(fx), y0f = floorf(fy);
        int   x0  = (int)x0f,   y0  = (int)y0f;
        float wx1 = fx - x0f,   wy1 = fy - y0f;   // (x-x0), (y-y0)
        float wx0 = 1.0f - wx1, wy0 = 1.0f - wy1; // (x1-x), (y1-y)

        bool vx0 = (x0 >= 0) && (x0 <= w - 1);
        bool vx1 = (x0 + 1 >= 0) && (x0 + 1 <= w - 1);
        bool vy0 = (y0 >= 0) && (y0 <= h - 1);
        bool vy1 = (y0 + 1 >= 0) && (y0 + 1 <= h - 1);
        int xc0 = min(max(x0, 0), w - 1),     xc1 = min(max(x0 + 1, 0), w - 1);
        int yc0 = min(max(y0, 0), h - 1),     yc1 = min(max(y0 + 1, 0), h - 1);

        const float* tgf = (j == 0) ? tgt0_full : tgt1_full;
        const float* tgl = (j == 0) ? tgt0_lvl  : tgt1_lvl;
        float m = mask[(((size_t)b * NBF + j) * h + y) * w + x];

#pragma unroll
        for (int c = 0; c < 3; ++c) {
            float va, vb, vc, vd;
            if (LVL == 0) {
                const float* base = tgf + (size_t)(b * 3 + c) * HH * WW;
                va = (vx0 && vy0) ? base[(size_t)yc0 * WW + xc0] : 0.0f;
                vb = (vx0 && vy1) ? base[(size_t)yc1 * WW + xc0] : 0.0f;
                vc = (vx1 && vy0) ? base[(size_t)yc0 * WW + xc1] : 0.0f;
                vd = (vx1 && vy1) ? base[(size_t)yc1 * WW + xc1] : 0.0f;
            } else if (PYR) {
                const float* base = tgl + (size_t)(b * 3 + c) * h * w;
                va = (vx0 && vy0) ? base[(size_t)yc0 * w + xc0] : 0.0f;
                vb = (vx0 && vy1) ? base[(size_t)yc1 * w + xc0] : 0.0f;
                vc = (vx1 && vy0) ? base[(size_t)yc0 * w + xc1] : 0.0f;
                vd = (vx1 && vy1) ? base[(size_t)yc1 * w + xc1] : 0.0f;
            } else {
                const float* base = tgf + (size_t)(b * 3 + c) * HH * WW;
                va = (vx0 && vy0) ? area_mean<S>(base, yc0, xc0) : 0.0f;
                vb = (vx0 && vy1) ? area_mean<S>(base, yc1, xc0) : 0.0f;
                vc = (vx1 && vy0) ? area_mean<S>(base, yc0, xc1) : 0.0f;
                vd = (vx1 && vy1) ? area_mean<S>(base, yc1, xc1) : 0.0f;
            }
            float wf = va * wx0 * wy0 + vb * wx0 * wy1 + vc * wx1 * wy0 + vd * wx1 * wy1;
            lsum += m * fabsf(rr[c] - wf);
        }
    }
    lsum *= scale;

    // block tree reduction in LDS, one float atomic per block
    __shared__ float sh[256];
    sh[threadIdx.x] = lsum;
    __syncthreads();
#pragma unroll
    for (int off = 128; off > 0; off >>= 1) {
        if (threadIdx.x < off) sh[threadIdx.x] += sh[threadIdx.x + off];
        __syncthreads();
    }
    if (threadIdx.x == 0)
        atomicAdd(&partials[blockIdx.x & (NPART - 1)], sh[0]);
}

// ---------------- final reduction via WMMA (ones^T x B accumulate) ----------------
// One wave32 block, EXEC all ones, uniform loop. Each v_wmma_f32_16x16x4_f32 with
// an all-ones A folds 64 partials into the f32 accumulator:
//   D[m,n] += sum_k B[k,n]  =>  sum over all 256 accum entries = 16 * total.
__global__ void reduce_wmma_kernel(const float* __restrict__ partials,
                                   float* __restrict__ out) {
    int lane = threadIdx.x;                 // 0..31
    v2f a = {1.0f, 1.0f};                   // all-ones A (layout-independent)
    v8f acc = {};
    for (int base = 0; base < NPART; base += 64) {
        v2f bm;
        bm[0] = partials[base + lane];
        bm[1] = partials[base + 32 + lane];
        acc = __builtin_amdgcn_wmma_f32_16x16x4_f32(
            /*neg_a=*/false, a, /*neg_b=*/false, bm,
            /*c_mod=*/(short)0, acc, /*reuse_a=*/false, /*reuse_b=*/false);
    }
    float v = acc[0] + acc[1] + acc[2] + acc[3] + acc[4] + acc[5] + acc[6] + acc[7];
#pragma unroll
    for (int off = 16; off > 0; off >>= 1)
        v += __shfl_xor(v, off, 32);
    if (lane == 0) out[0] = v * (1.0f / 16.0f);
}

// ---------------- host launch ----------------
extern "C" void kernel_launch(void* const* d_in, const int* in_sizes, int n_in,
                              void* d_out, int out_size, void* d_ws, size_t ws_size,
                              hipStream_t stream) {
    const float* ref      = (const float*)d_in[0];
    const float* depth[4] = {(const float*)d_in[1], (const float*)d_in[2],
                             (const float*)d_in[3], (const float*)d_in[4]};
    const float* mask[4]  = {(const float*)d_in[5], (const float*)d_in[6],
                             (const float*)d_in[7], (const float*)d_in[8]};
    const float* tgt[2]   = {(const float*)d_in[9], (const float*)d_in[10]};
    const float* pose     = (const float*)d_in[11];
    float* out = (float*)d_out;

    float* ws       = (float*)d_ws;
    float* partials = ws;                 // NPART floats
    float* rt       = ws + NPART;         // 16 * 12 floats
    float* pyr      = ws + NPART + 256;   // target pyramids

    const size_t N1 = (size_t)BB * CCH * (HH / 2) * (WW / 2);
    const size_t N2 = (size_t)BB * CCH * (HH / 4) * (WW / 4);
    const size_t N3 = (size_t)BB * CCH * (HH / 8) * (WW / 8);
    const size_t perimg = N1 + N2 + N3;
    const size_t needed = (size_t)(NPART + 256 + 2 * perimg) * sizeof(float);
    const bool use_pyr = (ws_size >= needed);

    zero_kernel<<<NPART / 256, 256, 0, stream>>>(partials);
    rot_kernel<<<1, 32, 0, stream>>>(pose, rt);

    float* lvlp[2][4] = {{nullptr, nullptr, nullptr, nullptr},
                         {nullptr, nullptr, nullptr, nullptr}};
    if (use_pyr) {
        for (int j = 0; j < 2; ++j) {
            float* p = pyr + (size_t)j * perimg;
            lvlp[j][1] = p;
            lvlp[j][2] = p + N1;
            lvlp[j][3] = p + N1 + N2;
            down_kernel<2><<<(int)(N1 / 256), 256, 0, stream>>>(tgt[j], lvlp[j][1], HH / 2, WW / 2);
            down_kernel<4><<<(int)(N2 / 256), 256, 0, stream>>>(tgt[j], lvlp[j][2], HH / 4, WW / 4);
            down_kernel<8><<<(int)(N3 / 256), 256, 0, stream>>>(tgt[j], lvlp[j][3], HH / 8, WW / 8);
        }
    }

    // level 0 (reads full-res targets directly)
    {
        int blocks = BB * HH * WW / 256;
        photo_kernel<0, true><<<blocks, 256, 0, stream>>>(
            ref, depth[0], mask[0], tgt[0], tgt[1], nullptr, nullptr, rt, partials);
    }
    // levels 1..3
    if (use_pyr) {
        photo_kernel<1, true><<<BB * (HH / 2) * (WW / 2) / 256, 256, 0, stream>>>(
            ref, depth[1], mask[1], tgt[0], tgt[1], lvlp[0][1], lvlp[1][1], rt, partials);
        photo_kernel<2, true><<<BB * (HH / 4) * (WW / 4) / 256, 256, 0, stream>>>(
            ref, depth[2], mask[2], tgt[0], tgt[1], lvlp[0][2], lvlp[1][2], rt, partials);
        photo_kernel<3, true><<<BB * (HH / 8) * (WW / 8) / 256, 256, 0, stream>>>(
            ref, depth[3], mask[3], tgt[0], tgt[1], lvlp[0][3], lvlp[1][3], rt, partials);
    } else {
        photo_kernel<1, false><<<BB * (HH / 2) * (WW / 2) / 256, 256, 0, stream>>>(
            ref, depth[1], mask[1], tgt[0], tgt[1], nullptr, nullptr, rt, partials);
        photo_kernel<2, false><<<BB * (HH / 4) * (WW / 4) / 256, 256, 0, stream>>>(
            ref, depth[2], mask[2], tgt[0], tgt[1], nullptr, nullptr, rt, partials);
        photo_kernel<3, false><<<BB * (HH / 8) * (WW / 8) / 256, 256, 0, stream>>>(
            ref, depth[3], mask[3], tgt[0], tgt[1], nullptr, nullptr, rt, partials);
    }

    reduce_wmma_kernel<<<1, 32, 0, stream>>>(partials, out);
}